// graph_attention_13640816132600
// MI455X (gfx1250) — compile-verified
//
#include <hip/hip_runtime.h>
#include <math.h>
#include <stdint.h>

#define B_    128
#define N_    256
#define D_    10
#define ROWS  32          // rows of the rel matrix per block
#define DIS_THRE 30.0f

// Async global->LDS staging (gfx1250 ASYNCcnt-tracked path) via inline asm:
// the clang builtin's pointer params use the cuda_device/cuda_shared LangAS,
// which cannot be spelled from HIP source, so we emit the ISA directly.
#if defined(__gfx1250__)
#define USE_ASYNC_LDS 1
#else
#define USE_ASYNC_LDS 0
#endif

__global__ __launch_bounds__(256) void graph_attention_kernel(
    const float* __restrict__ X,   // (B,N,10)
    const float* __restrict__ w1,  // (4,2)
    const float* __restrict__ b1,  // (4,)
    const float* __restrict__ w2,  // (1,4)
    const float* __restrict__ b2,  // (1,)
    const float* __restrict__ wf,  // (1,10)
    const float* __restrict__ bf,  // (1,)
    const float* __restrict__ wv,  // (1,1)
    const float* __restrict__ bv,  // (1,)
    float* __restrict__ deg_out,   // (B,N) written as float
    float* __restrict__ rel_out)   // (B,N,N)
{
    __shared__ __align__(16) float xraw[N_ * D_];   // raw X[b] slice, 10240 B
    __shared__ float2 spos[N_];        // (x,y)   = X[...,1:3]
    __shared__ float2 svel[N_];        // (vx,vy) = X[...,3:5]
    __shared__ float2 sdsi[N_];        // (dir, si)
    __shared__ float  ssj [N_];        // sj

    const int t     = threadIdx.x;
    const int b     = blockIdx.x >> 3;            // 8 row-tiles per batch
    const int nbase = (blockIdx.x & 7) * ROWS;
    const float* Xb = X + (size_t)b * (N_ * D_);

    // ---- stage X[b] (2560 floats = 640 x b128) into LDS ----
#if USE_ASYNC_LDS
    #pragma unroll
    for (int i = 0; i < 3; ++i) {
        const int idx = t + 256 * i;              // 16-byte chunk index
        if (idx < (N_ * D_ / 4)) {
            const float* gp = Xb + idx * 4;
            // generic address of a __shared__ object = {aperture, lds_offset};
            // low 32 bits are the DS byte offset (ISA flat->LDS mapping).
            unsigned lds_addr = (unsigned)(uintptr_t)(const void*)(xraw + idx * 4);
            asm volatile("global_load_async_to_lds_b128 %0, %1, off"
                         :: "v"(lds_addr), "v"(gp)
                         : "memory");
        }
    }
    asm volatile("s_wait_asynccnt 0" ::: "memory");
#else
    for (int i = t; i < N_ * D_; i += 256) xraw[i] = Xb[i];
#endif
    __syncthreads();

    // ---- uniform weights (scalar loads) ----
    float w1k0[4], w1k1[4], b1v[4], w2v[4];
    #pragma unroll
    for (int k = 0; k < 4; ++k) {
        w1k0[k] = w1[2 * k];
        w1k1[k] = w1[2 * k + 1];
        b1v[k]  = b1[k];
        w2v[k]  = w2[k];
    }
    const float b2v = b2[0], bfv = bf[0], wvv = wv[0], bvv = bv[0];
    float wfa[5], wfb[5];
    #pragma unroll
    for (int i = 0; i < 5; ++i) { wfa[i] = wf[i]; wfb[i] = wf[5 + i]; }

    // ---- per-point derived quantities into SoA LDS ----
    {
        const float* xp = xraw + t * D_;
        const float x0 = xp[0], x1 = xp[1], x2 = xp[2], x3 = xp[3], x4 = xp[4];
        const float x5 = xp[5], x6 = xp[6], x7 = xp[7], x8 = xp[8], x9 = xp[9];
        spos[t] = make_float2(x1, x2);
        svel[t] = make_float2(x3, x4);
        // feat = [x0, x5, x6, x7, x9]
        const float sj = x0*wfa[0] + x5*wfa[1] + x6*wfa[2] + x7*wfa[3] + x9*wfa[4];
        const float si = x0*wfb[0] + x5*wfb[1] + x6*wfb[2] + x7*wfb[3] + x9*wfb[4];
        sdsi[t] = make_float2(x8, si);
        ssj[t]  = sj;
    }
    __syncthreads();

    // ---- compute: one wave32 per output row; lane handles m = lane + 32k ----
    const int wave = t >> 5;
    const int lane = t & 31;

    #pragma unroll
    for (int r = 0; r < ROWS / 8; ++r) {
        const int n = nbase + wave * (ROWS / 8) + r;
        const float2 pn  = spos[n];
        const float2 vn  = svel[n];
        const float  dn  = sdsi[n].x;
        const float  sjn = ssj[n];
        float an[4];
        #pragma unroll
        for (int k = 0; k < 4; ++k) an[k] = fmaf(dn, w1k0[k], b1v[k]);

        float relv[8];
        float rmax = -INFINITY;
        #pragma unroll
        for (int k = 0; k < 8; ++k) {
            const int m = lane + 32 * k;
            const float2 pm = spos[m];
            const float2 vm = svel[m];
            const float2 ds_m = sdsi[m];

            const float dx  = pn.x - pm.x, dy = pn.y - pm.y;
            const float dis = __builtin_sqrtf(fmaf(dx, dx, fmaf(dy, dy, 1e-8f)));
            const float ex  = vn.x - vm.x, ey = vn.y - vm.y;
            const float vel = __builtin_sqrtf(fmaf(ex, ex, fmaf(ey, ey, 1e-8f)));

            const float dm = ds_m.x;
            // relu(pre) then softmax over 4, dot w2
            const float t0 = fmaxf(fmaf(dm, w1k1[0], an[0]), 0.0f);
            const float t1 = fmaxf(fmaf(dm, w1k1[1], an[1]), 0.0f);
            const float t2 = fmaxf(fmaf(dm, w1k1[2], an[2]), 0.0f);
            const float t3 = fmaxf(fmaf(dm, w1k1[3], an[3]), 0.0f);
            const float e0 = __expf(t0), e1 = __expf(t1);
            const float e2 = __expf(t2), e3 = __expf(t3);
            const float hs  = (e0 + e1) + (e2 + e3);
            const float num = fmaf(e0, w2v[0], fmaf(e1, w2v[1], fmaf(e2, w2v[2], e3 * w2v[3])));
            const float dscore = num / hs + b2v;

            const float fscore = sjn + ds_m.y + bfv;
            const float vscore = fmaf(wvv, vel, bvv);

            const float raw = (DIS_THRE / (dis + 1e-4f)) * dscore * fscore * vscore;
            const float v   = (dis > DIS_THRE) ? -INFINITY : raw;
            relv[k] = v;
            rmax = fmaxf(rmax, v);
        }

        // wave32 max reduction
        #pragma unroll
        for (int o = 16; o > 0; o >>= 1) rmax = fmaxf(rmax, __shfl_xor(rmax, o, 32));

        float ev[8];
        float s = 0.0f;
        #pragma unroll
        for (int k = 0; k < 8; ++k) { ev[k] = __expf(relv[k] - rmax); s += ev[k]; }
        #pragma unroll
        for (int o = 16; o > 0; o >>= 1) s += __shfl_xor(s, o, 32);
        const float inv = 1.0f / s;

        int cnt = 0;
        float* __restrict__ row = rel_out + ((size_t)(b * N_ + n)) * N_;
        #pragma unroll
        for (int k = 0; k < 8; ++k) {
            const float o_ = ev[k] * inv;
            cnt += (o_ != 0.0f) ? 1 : 0;
            row[lane + 32 * k] = o_;     // coalesced 128B per store
        }
        #pragma unroll
        for (int o = 16; o > 0; o >>= 1) cnt += __shfl_xor(cnt, o, 32);
        if (lane == 0) deg_out[b * N_ + n] = (float)(cnt - 1);
    }
}

extern "C" void kernel_launch(void* const* d_in, const int* in_sizes, int n_in,
                              void* d_out, int out_size, void* d_ws, size_t ws_size,
                              hipStream_t stream) {
    (void)in_sizes; (void)n_in; (void)out_size; (void)d_ws; (void)ws_size;
    const float* X  = (const float*)d_in[0];
    const float* w1 = (const float*)d_in[1];
    const float* b1 = (const float*)d_in[2];
    const float* w2 = (const float*)d_in[3];
    const float* b2 = (const float*)d_in[4];
    const float* wf = (const float*)d_in[5];
    const float* bf = (const float*)d_in[6];
    const float* wv = (const float*)d_in[7];
    const float* bv = (const float*)d_in[8];

    float* deg = (float*)d_out;              // B*N floats
    float* rel = (float*)d_out + B_ * N_;    // B*N*N floats

    dim3 grid(B_ * (N_ / ROWS));             // 1024 blocks
    dim3 block(256);                         // 8 wave32
    graph_attention_kernel<<<grid, block, 0, stream>>>(
        X, w1, b1, w2, b2, wf, bf, wv, bv, deg, rel);
}